// ClusterModel_7121055776890
// MI455X (gfx1250) — compile-verified
//
#include <hip/hip_runtime.h>

// ---------------------------------------------------------------------------
// ClusterModel fused pipeline for gfx1250 (MI455X)
//   acc_sum = SpMM(LI, ebs)            [N,64] f32  (scatter atomics, L2-resident)
//   acc_dot = SpMM(L,  ebs)            [N,64] f32
//   out     = leaky_relu(acc_sum @ W_side + (acc_dot * entity) @ W_dot)
//
// Roofline: ~100 MB of HBM streaming (~5us at 23.3 TB/s); ebs (25.6 MB) and
// both accumulators (51 MB) are L2-resident (192 MB L2), so the limiter is
// L2 atomic throughput -> use native GLOBAL_ATOMIC_ADD_F32 (unsafeAtomicAdd),
// never a CAS loop. Dense stage is 1.6 GFLOP -> exact-fp32 WMMA, no precision
// downgrade needed.
// ---------------------------------------------------------------------------

typedef __attribute__((ext_vector_type(2))) float v2f;
typedef __attribute__((ext_vector_type(8))) float v8f;

#define D 64
#define LEAKY 0.2f

// ---- Kernel 1: zero the accumulators (float4-wide) ------------------------
__global__ __launch_bounds__(256) void zero_f4(float4* __restrict__ p, int n4) {
    int i = blockIdx.x * 256 + threadIdx.x;
    if (i < n4) p[i] = make_float4(0.f, 0.f, 0.f, 0.f);
}

// ---- Kernel 2: COO scatter SpMM: acc[row] += val * ebs[col] ---------------
// 16 lanes per edge; each lane does one coalesced float4 gather of ebs[col]
// (L2-resident) and 4 native f32 global atomics (GLOBAL_ATOMIC_ADD_F32,
// no-return, STOREcnt-tracked). Edge scalars are broadcast loads (16 lanes
// hit the same address -> single transaction).
__global__ __launch_bounds__(256) void spmm_scatter(
    const int* __restrict__ rows, const int* __restrict__ cols,
    const float* __restrict__ vals, const float* __restrict__ dense,
    float* __restrict__ acc, int E)
{
    long t = (long)blockIdx.x * 256 + threadIdx.x;
    int e = (int)(t >> 4);
    if (e >= E) return;
    int c4 = ((int)t & 15) * 4;

    int   r = rows[e];
    int   c = cols[e];
    float v = vals[e];

    const float4 g = *(const float4*)(dense + (long)c * D + c4);
    float* dst = acc + (long)r * D + c4;
    unsafeAtomicAdd(dst + 0, v * g.x);
    unsafeAtomicAdd(dst + 1, v * g.y);
    unsafeAtomicAdd(dst + 2, v * g.z);
    unsafeAtomicAdd(dst + 3, v * g.w);
}

// ---- Kernel 3: fused dual-GEMM + leaky-ReLU via v_wmma_f32_16x16x4_f32 ----
// Block = 128 threads = 4 wave32s; block handles 16 output rows, wave w
// handles output columns [16w, 16w+16). 16 k-steps of K=4 cover K=64, with
// two WMMAs per step (sum path, dot path) into one f32 accumulator.
//
// Fragment layouts (CDNA5 ISA 7.12.2, wave32):
//   A 16x4 f32:  lanes 0-15 -> M=lane,     K = {0,1} in v0,v1
//                lanes 16-31 -> M=lane-16, K = {2,3} in v0,v1
//   B 4x16 f32:  lanes 0-15 -> N=lane,     K = {0,1} in v0,v1 (row-striped)
//                lanes 16-31 -> N=lane-16, K = {2,3}
//   C/D 16x16:   VGPR j -> M = j (lanes 0-15) or j+8 (lanes 16-31), N = lane&15
// EXEC is all-ones (no divergence; N % 16 == 0) as WMMA requires.
__global__ __launch_bounds__(128) void fused_gemm_lrelu(
    const float* __restrict__ acc_sum,   // [N,64]
    const float* __restrict__ acc_dot,   // [N,64]
    const float* __restrict__ entity,    // [N,64]
    const float* __restrict__ Wside,     // [64,64] row-major
    const float* __restrict__ Wdot,      // [64,64] row-major
    float* __restrict__ out)             // [N,64]
{
    const int tile = blockIdx.x;          // 16-row tile of output
    const int wave = threadIdx.x >> 5;    // 0..3 -> 16-col tile
    const int lane = threadIdx.x & 31;
    const int r    = lane & 15;           // A row within tile / B,D col within tile
    const int h    = lane >> 4;           // K half-select
    const int n    = (lane & 15) + wave * 16;  // B/D column

    const long row = (long)tile * 16 + r;
    const float* Ss = acc_sum + row * D;
    const float* Sd = acc_dot + row * D;
    const float* Ee = entity  + row * D;

    v8f c = {};
#pragma unroll
    for (int k = 0; k < 16; ++k) {
        const int kk = k * 4 + h * 2;     // this lane's first K index this step

        // --- sum path: A = acc_sum, B = W_side ---
        v2f a1; a1.x = Ss[kk];  a1.y = Ss[kk + 1];
        v2f b1; b1.x = Wside[kk * D + n]; b1.y = Wside[(kk + 1) * D + n];
        c = __builtin_amdgcn_wmma_f32_16x16x4_f32(
                false, a1, false, b1, (short)0, c, false, false);

        // --- dot path: A = acc_dot * entity, B = W_dot ---
        v2f a2; a2.x = Sd[kk] * Ee[kk]; a2.y = Sd[kk + 1] * Ee[kk + 1];
        v2f b2; b2.x = Wdot[kk * D + n]; b2.y = Wdot[(kk + 1) * D + n];
        c = __builtin_amdgcn_wmma_f32_16x16x4_f32(
                false, a2, false, b2, (short)0, c, false, false);
    }

    // Epilogue: leaky ReLU + store per C/D VGPR layout.
    float* o = out + ((long)tile * 16) * D + wave * 16 + (lane & 15);
#pragma unroll
    for (int j = 0; j < 8; ++j) {
        const int m = j + h * 8;
        float v = c[j];
        v = (v > 0.f) ? v : LEAKY * v;
        o[(long)m * D] = v;
    }
}

// ---------------------------------------------------------------------------
extern "C" void kernel_launch(void* const* d_in, const int* in_sizes, int n_in,
                              void* d_out, int out_size, void* d_ws, size_t ws_size,
                              hipStream_t stream)
{
    const float* ebs      = (const float*)d_in[0];
    const float* entity   = (const float*)d_in[1];
    const float* W_side   = (const float*)d_in[2];
    const float* W_dot    = (const float*)d_in[3];
    const float* LI_vals  = (const float*)d_in[4];
    const float* L_vals   = (const float*)d_in[5];
    const int*   LI_rows  = (const int*)d_in[6];
    const int*   LI_cols  = (const int*)d_in[7];
    const int*   L_rows   = (const int*)d_in[8];
    const int*   L_cols   = (const int*)d_in[9];
    float*       out      = (float*)d_out;

    const int N = in_sizes[0] / D;   // 100000
    const int E = in_sizes[4];       // 1000000

    float* acc_sum = (float*)d_ws;              // [N,64]
    float* acc_dot = acc_sum + (long)N * D;     // [N,64]

    // 1) zero accumulators (2*N*64 floats)
    {
        int n4 = (2 * N * D) / 4;
        int blocks = (n4 + 255) / 256;
        zero_f4<<<blocks, 256, 0, stream>>>((float4*)d_ws, n4);
    }

    // 2) scatter SpMMs (16 lanes per edge)
    {
        long threads = (long)E * 16;
        int blocks = (int)((threads + 255) / 256);
        spmm_scatter<<<blocks, 256, 0, stream>>>(LI_rows, LI_cols, LI_vals,
                                                 ebs, acc_sum, E);
        spmm_scatter<<<blocks, 256, 0, stream>>>(L_rows, L_cols, L_vals,
                                                 ebs, acc_dot, E);
    }

    // 3) fused dual-GEMM + leaky ReLU (N % 16 == 0)
    {
        int blocks = N / 16;
        fused_gemm_lrelu<<<blocks, 128, 0, stream>>>(acc_sum, acc_dot, entity,
                                                     W_side, W_dot, out);
    }
}